// CTC_22866405884308
// MI455X (gfx1250) — compile-verified
//
#include <hip/hip_runtime.h>
#include <hip/hip_bf16.h>
#include <stdint.h>

// Problem constants from setup_inputs(): B=32, T=1000, D=4096, L=100.
#define CB 32
#define CT 1000
#define CD 4096
#define CL 100
#define CS 201            // 2*L+1
#define NEGV -1e30f

__device__ __forceinline__ float lae(float a, float b) {
    // logaddexp, safe for a==b==-1e30 (matches jnp.logaddexp behavior in f32)
    float m = fmaxf(a, b);
    float d = fabsf(a - b);
    return m + log1pf(__expf(-d));
}

// ---------------------------------------------------------------------------
// Kernel 1: per-row logsumexp over D=4096 + gather of extended-label logits.
// One workgroup per (b,t) row. The 16KB row is staged into LDS via gfx1250
// async-to-LDS loads (ASYNCcnt path) with NT hint (stream is 524MB > L2).
// ---------------------------------------------------------------------------
__global__ __launch_bounds__(256) void k_rowlse(const float* __restrict__ hs,
                                                const int* __restrict__ ys,
                                                float* __restrict__ lpext) {
    __shared__ __align__(16) float buf[CD];   // full row resident in LDS
    __shared__ float red[256];
    __shared__ float s_lse;

    const int row = blockIdx.x;               // row = b*T + t
    const int b   = row / CT;
    const int tid = threadIdx.x;
    const float* rp = hs + (size_t)row * CD;

    // LDS byte offset of buf: generic shared addr low 32 bits == LDS offset.
    uint32_t lbase = (uint32_t)(uintptr_t)(&buf[0]);

    // 1024 x b128 chunks, 4 per thread: each wave instr moves 32x16B = 512B.
#pragma unroll
    for (int k = 0; k < 4; ++k) {
        int c = tid + 256 * k;
        unsigned long long ga = (unsigned long long)(uintptr_t)(rp + 4 * c);
        uint32_t la = lbase + 16u * (uint32_t)c;
        asm volatile("global_load_async_to_lds_b128 %0, %1, off th:TH_LOAD_NT"
                     :: "v"(la), "v"(ga) : "memory");
    }
    asm volatile("s_wait_asynccnt 0" ::: "memory");
    __syncthreads();                          // all waves' async data landed

    // Pull 16 values/thread from LDS into registers; keep buf intact for gather.
    const float4* bv = (const float4*)buf;
    float4 v[4];
    float m = NEGV;
#pragma unroll
    for (int k = 0; k < 4; ++k) {
        v[k] = bv[tid + 256 * k];
        m = fmaxf(m, fmaxf(fmaxf(v[k].x, v[k].y), fmaxf(v[k].z, v[k].w)));
    }
    red[tid] = m;
    __syncthreads();
    for (int off = 128; off > 0; off >>= 1) {
        if (tid < off) red[tid] = fmaxf(red[tid], red[tid + off]);
        __syncthreads();
    }
    m = red[0];
    __syncthreads();                          // protect red[] before reuse

    float ssum = 0.f;
#pragma unroll
    for (int k = 0; k < 4; ++k) {
        ssum += __expf(v[k].x - m) + __expf(v[k].y - m) +
                __expf(v[k].z - m) + __expf(v[k].w - m);
    }
    red[tid] = ssum;
    __syncthreads();
    for (int off = 128; off > 0; off >>= 1) {
        if (tid < off) red[tid] += red[tid + off];
        __syncthreads();
    }
    if (tid == 0) s_lse = m + __logf(red[0]);
    __syncthreads();

    // Gather extended labels from the LDS-resident row (avoids scattered HBM
    // pulls in the DP kernel). ext[s] = blank(0) if s even else ys[b][s>>1].
    if (tid < CS) {
        int label = (tid & 1) ? ys[b * CL + (tid >> 1)] : 0;
        lpext[(size_t)row * CS + tid] = buf[label] - s_lse;
    }
}

// ---------------------------------------------------------------------------
// Kernel 2: CTC forward DP. One block per batch element, one state per lane,
// double-buffered alpha in LDS, 1 barrier per timestep. Register software
// pipeline + global_prefetch_b8 hide L2 latency of the per-step lp row.
// ---------------------------------------------------------------------------
__global__ __launch_bounds__(256) void k_ctc(const float* __restrict__ lpext,
                                             const int* __restrict__ hlen,
                                             const int* __restrict__ ys,
                                             const int* __restrict__ yl,
                                             float* __restrict__ lossb) {
    __shared__ float abuf[2][CS + 7];
    const int b = blockIdx.x;
    const int s = threadIdx.x;
    const bool act = (s < CS);
    int Tb = hlen[b];
    Tb = (Tb < 1) ? 1 : (Tb > CT ? CT : Tb);
    const float* lpb = lpext + (size_t)b * CT * CS;

    // skip (s-2 -> s) allowed iff s odd, s>=3, and adjacent labels differ
    bool skip_ok = false;
    if (act && s >= 3 && (s & 1))
        skip_ok = (ys[b * CL + (s >> 1)] != ys[b * CL + (s >> 1) - 1]);

    if (act) abuf[0][s] = (s <= 1) ? lpb[s] : NEGV;
    __syncthreads();

    int cur = 0;
    float lp_next = (act && Tb > 1) ? lpb[CS + s] : 0.f;
    for (int t = 1; t < Tb; ++t) {
        float lp = lp_next;
        if (act && t + 1 < Tb) lp_next = lpb[(size_t)(t + 1) * CS + s];
        if (act && t + 4 < Tb)
            __builtin_prefetch(&lpb[(size_t)(t + 4) * CS + s], 0, 0);

        float nv = 0.f;
        if (act) {
            float x0 = abuf[cur][s];
            float x1 = (s >= 1) ? abuf[cur][s - 1] : NEGV;
            float x2 = skip_ok ? abuf[cur][s - 2] : NEGV;
            nv = lae(lae(x0, x1), x2) + lp;
        }
        if (act) abuf[cur ^ 1][s] = nv;
        __syncthreads();                      // one barrier per step
        cur ^= 1;
    }

    if (s == 0) {
        int tl = yl[b];
        float e1 = abuf[cur][2 * tl];
        float e2 = abuf[cur][2 * tl - 1];
        lossb[b] = lae(e1, e2);
    }
}

// ---------------------------------------------------------------------------
// Kernel 3: final scalar: loss = -sum_b(lossb) / B
// ---------------------------------------------------------------------------
__global__ void k_final(const float* __restrict__ lossb, float* __restrict__ out) {
    if (threadIdx.x == 0) {
        float ssum = 0.f;
        for (int i = 0; i < CB; ++i) ssum += lossb[i];
        out[0] = -ssum / (float)CB;
    }
}

extern "C" void kernel_launch(void* const* d_in, const int* in_sizes, int n_in,
                              void* d_out, int out_size, void* d_ws, size_t ws_size,
                              hipStream_t stream) {
    const float* hs   = (const float*)d_in[0];  // (B,T,D) f32
    const int*   hlen = (const int*)  d_in[1];  // (B,)
    const int*   ys   = (const int*)  d_in[2];  // (B,L)
    const int*   yl   = (const int*)  d_in[3];  // (B,)

    float* lpext = (float*)d_ws;                        // B*T*S floats (~25.7MB)
    float* lossb = lpext + (size_t)CB * CT * CS;        // B floats

    k_rowlse<<<CB * CT, 256, 0, stream>>>(hs, ys, lpext);
    k_ctc<<<CB, 256, 0, stream>>>(lpext, hlen, ys, yl, lossb);
    k_final<<<1, 32, 0, stream>>>(lossb, (float*)d_out);
}